// GCNBaseline_60619168416467
// MI455X (gfx1250) — compile-verified
//
#include <hip/hip_runtime.h>
#include <hip/hip_bf16.h>
#include <math.h>

typedef __attribute__((ext_vector_type(2))) float v2f;
typedef __attribute__((ext_vector_type(8))) float v8f;

#define N_NODES   100000
#define NCOL      64           // HIDDEN
#define N_FEAT    128
#define N_GRAPHS  512
#define N_CLASSES 3
#define BN_EPS    1e-5f

// ---------------- degree / norm ----------------
__global__ __launch_bounds__(256) void init_deg_kernel(float* deg, int n) {
    int i = blockIdx.x * blockDim.x + threadIdx.x;
    if (i < n) deg[i] = 1.0f;   // self-loop contributes 1
}

__global__ __launch_bounds__(256) void edge_deg_kernel(const int* __restrict__ dst,
                                                       float* __restrict__ deg, int nE) {
    int e = blockIdx.x * blockDim.x + threadIdx.x;
    if (e < nE) atomicAdd(&deg[dst[e]], 1.0f);
}

__global__ __launch_bounds__(256) void dinv_kernel(float* deg, int n) {
    int i = blockIdx.x * blockDim.x + threadIdx.x;
    if (i < n) deg[i] = rsqrtf(deg[i]);   // deg >= 1 always
}

// ---------------- fp32 WMMA GEMM: Y[rows,64] = X[rows,K] @ W[K,64] ----------------
// One wave -> 16 rows x 64 cols (4 accumulators). K stepped by 4 (V_WMMA_F32_16X16X4_F32).
// A frag (16x4 f32): lanes 0-15 -> M=lane, K = k0+{0,1}; lanes 16-31 -> M=lane-16, K = k0+{2,3}
// B frag (4x16 f32): lanes 0-15 -> N=lane, K = k0+{0,1}; lanes 16-31 -> N=lane-16, K = k0+{2,3}
// C/D (16x16 f32): vgpr j, lanes 0-15 -> (M=j, N=lane); lanes 16-31 -> (M=j+8, N=lane-16)
//
// W staged to LDS in K-pair-interleaved layout: sW[k>>1][n][k&1], so a lane's
// B fragment {W[k][n], W[k+1][n]} is a single contiguous 8B word -> ds_load_b64,
// no repacking movs before the WMMAs.
__global__ __launch_bounds__(256) void gemm_wmma_kernel(const float* __restrict__ X,
                                                        const float* __restrict__ W,
                                                        float* __restrict__ Y,
                                                        int nRowTiles, int K) {
    __shared__ float sW[N_FEAT * NCOL];   // up to 128*64 = 32 KB
    for (int i = threadIdx.x; i < K * NCOL; i += blockDim.x) {
        const int k = i >> 6;          // row in W
        const int n = i & 63;          // col in W
        sW[(k >> 1) * (NCOL * 2) + n * 2 + (k & 1)] = W[i];
    }
    __syncthreads();

    const int lane = threadIdx.x & 31;
    const int wave = threadIdx.x >> 5;
    const int tile = blockIdx.x * (blockDim.x >> 5) + wave;   // wave-uniform
    if (tile >= nRowTiles) return;                            // uniform branch: EXEC stays all-1

    const int row0  = tile * 16;
    const int m     = lane & 15;
    const int half  = lane >> 4;          // 0 for lanes 0-15, 1 for lanes 16-31
    const int khalf = half << 1;          // 0 or 2

    v8f acc0 = {}, acc1 = {}, acc2 = {}, acc3 = {};
    const float* __restrict__ xrow = X + (size_t)(row0 + m) * K;
    // per-lane base into paired-W LDS: (kpair = half) row + column m
    const float* __restrict__ wbase = sW + half * (NCOL * 2) + m * 2;

    for (int k0 = 0; k0 < K; k0 += 4) {
        v2f a;
        a.x = xrow[k0 + khalf];
        a.y = xrow[k0 + khalf + 1];
        const float* wrow = wbase + (k0 >> 1) * (NCOL * 2);
        const v2f b0 = *(const v2f*)(wrow);
        const v2f b1 = *(const v2f*)(wrow + 32);
        const v2f b2 = *(const v2f*)(wrow + 64);
        const v2f b3 = *(const v2f*)(wrow + 96);
        acc0 = __builtin_amdgcn_wmma_f32_16x16x4_f32(false, a, false, b0, (short)0, acc0, false, false);
        acc1 = __builtin_amdgcn_wmma_f32_16x16x4_f32(false, a, false, b1, (short)0, acc1, false, false);
        acc2 = __builtin_amdgcn_wmma_f32_16x16x4_f32(false, a, false, b2, (short)0, acc2, false, false);
        acc3 = __builtin_amdgcn_wmma_f32_16x16x4_f32(false, a, false, b3, (short)0, acc3, false, false);
    }

    const int rbase = row0 + (half << 3);   // +8 for upper half-wave
    #pragma unroll
    for (int j = 0; j < 8; ++j) {
        float* yr = Y + (size_t)(rbase + j) * NCOL + m;
        yr[0]  = acc0[j];
        yr[16] = acc1[j];
        yr[32] = acc2[j];
        yr[48] = acc3[j];
    }
}

// ---------------- edge aggregation: out[d] += h[s] * dinv[s]*dinv[d] ----------------
// wave-per-edge; lane carries 2 floats (32 lanes * 8B = full 64-float row, coalesced)
__global__ __launch_bounds__(256) void aggregate_kernel(const float* __restrict__ H,
                                                        const float* __restrict__ dinv,
                                                        const int* __restrict__ src,
                                                        const int* __restrict__ dst,
                                                        float* __restrict__ out, int nE) {
    const int lane = threadIdx.x & 31;
    const int e = blockIdx.x * (blockDim.x >> 5) + (threadIdx.x >> 5);
    if (e >= nE) return;
    const int s = src[e];
    const int d = dst[e];
    const float nrm = dinv[s] * dinv[d];
    const float2 hv = ((const float2*)(H + (size_t)s * NCOL))[lane];
    float* o = out + (size_t)d * NCOL + lane * 2;
    atomicAdd(o,     hv.x * nrm);
    atomicAdd(o + 1, hv.y * nrm);
}

// ---------------- fused self-loop + bias + BN(eval) + ReLU, in place on agg ----------------
__global__ __launch_bounds__(256) void post_bn_relu_kernel(float* __restrict__ agg,
                                                           const float* __restrict__ H,
                                                           const float* __restrict__ dinv,
                                                           const float* __restrict__ b,
                                                           const float* __restrict__ g,
                                                           const float* __restrict__ bt,
                                                           const float* __restrict__ rm,
                                                           const float* __restrict__ rv,
                                                           int total) {
    int i = blockIdx.x * blockDim.x + threadIdx.x;
    if (i >= total) return;
    const int node = i >> 6;
    const int f    = i & 63;
    const float di = dinv[node];
    float v = agg[i] + H[i] * di * di + b[f];
    v = (v - rm[f]) * rsqrtf(rv[f] + BN_EPS) * g[f] + bt[f];
    agg[i] = fmaxf(v, 0.0f);
}

// ---------------- global mean pool: atomic accumulate ----------------
__global__ __launch_bounds__(256) void pool_accum_kernel(const float* __restrict__ H,
                                                         const int* __restrict__ batch,
                                                         float* __restrict__ sums,
                                                         float* __restrict__ cnts, int nNodes) {
    const int lane = threadIdx.x & 31;
    const int node = blockIdx.x * (blockDim.x >> 5) + (threadIdx.x >> 5);
    if (node >= nNodes) return;
    const int gidx = batch[node];
    const float2 hv = ((const float2*)(H + (size_t)node * NCOL))[lane];
    float* o = sums + (size_t)gidx * NCOL + lane * 2;
    atomicAdd(o,     hv.x);
    atomicAdd(o + 1, hv.y);
    if (lane == 0) atomicAdd(&cnts[gidx], 1.0f);
}

// ---------------- classifier: out[G,3] = (sums/cnt) @ Wc + bc ----------------
__global__ __launch_bounds__(256) void classify_kernel(const float* __restrict__ sums,
                                                       const float* __restrict__ cnts,
                                                       const float* __restrict__ Wc,
                                                       const float* __restrict__ bc,
                                                       float* __restrict__ out) {
    int gph = blockIdx.x * blockDim.x + threadIdx.x;
    if (gph >= N_GRAPHS) return;
    const float inv = 1.0f / fmaxf(cnts[gph], 1.0f);
    float a0 = bc[0], a1 = bc[1], a2 = bc[2];
    const float* srow = sums + (size_t)gph * NCOL;
    #pragma unroll 4
    for (int f = 0; f < NCOL; ++f) {
        const float p = srow[f] * inv;
        a0 += p * Wc[f * 3 + 0];
        a1 += p * Wc[f * 3 + 1];
        a2 += p * Wc[f * 3 + 2];
    }
    out[gph * 3 + 0] = a0;
    out[gph * 3 + 1] = a1;
    out[gph * 3 + 2] = a2;
}

extern "C" void kernel_launch(void* const* d_in, const int* in_sizes, int n_in,
                              void* d_out, int out_size, void* d_ws, size_t ws_size,
                              hipStream_t stream) {
    const float* x   = (const float*)d_in[0];
    const float* W1  = (const float*)d_in[1];
    const float* b1  = (const float*)d_in[2];
    const float* g1  = (const float*)d_in[3];
    const float* bt1 = (const float*)d_in[4];
    const float* rm1 = (const float*)d_in[5];
    const float* rv1 = (const float*)d_in[6];
    const float* W2  = (const float*)d_in[7];
    const float* b2  = (const float*)d_in[8];
    const float* g2  = (const float*)d_in[9];
    const float* bt2 = (const float*)d_in[10];
    const float* rm2 = (const float*)d_in[11];
    const float* rv2 = (const float*)d_in[12];
    const float* Wc  = (const float*)d_in[13];
    const float* bc  = (const float*)d_in[14];
    const int*   ei  = (const int*)d_in[15];   // [2, E] row-major
    const int*   bat = (const int*)d_in[16];

    const int nE = in_sizes[15] / 2;
    const int* srcE = ei;
    const int* dstE = ei + nE;

    // workspace carve-up (float offsets, 256-float aligned)
    float* ws = (float*)d_ws;
    float* dinv = ws;                       // 100000 (padded 100352)
    float* sums = ws + 100352;              // 512*64 = 32768
    float* cnts = ws + 100352 + 32768;      // 512 (padded 512)
    float* bufA = ws + 133632;              // 6,400,000  (gemm out)
    float* bufB = ws + 133632 + 6400000;    // 6,400,000  (agg / post out)

    const int total = N_NODES * NCOL;
    const int rowTiles = N_NODES / 16;      // 6250, exact
    const dim3 blk(256);

    // degrees -> dinv
    init_deg_kernel<<<(N_NODES + 255) / 256, blk, 0, stream>>>(dinv, N_NODES);
    edge_deg_kernel<<<(nE + 255) / 256, blk, 0, stream>>>(dstE, dinv, nE);
    dinv_kernel<<<(N_NODES + 255) / 256, blk, 0, stream>>>(dinv, N_NODES);

    // ---- layer 1 ----
    gemm_wmma_kernel<<<(rowTiles + 7) / 8, blk, 0, stream>>>(x, W1, bufA, rowTiles, N_FEAT);
    hipMemsetAsync(bufB, 0, (size_t)total * sizeof(float), stream);
    aggregate_kernel<<<(nE + 7) / 8, blk, 0, stream>>>(bufA, dinv, srcE, dstE, bufB, nE);
    post_bn_relu_kernel<<<(total + 255) / 256, blk, 0, stream>>>(bufB, bufA, dinv,
                                                                 b1, g1, bt1, rm1, rv1, total);

    // ---- layer 2 ----
    gemm_wmma_kernel<<<(rowTiles + 7) / 8, blk, 0, stream>>>(bufB, W2, bufA, rowTiles, NCOL);
    hipMemsetAsync(bufB, 0, (size_t)total * sizeof(float), stream);
    aggregate_kernel<<<(nE + 7) / 8, blk, 0, stream>>>(bufA, dinv, srcE, dstE, bufB, nE);
    post_bn_relu_kernel<<<(total + 255) / 256, blk, 0, stream>>>(bufB, bufA, dinv,
                                                                 b2, g2, bt2, rm2, rv2, total);

    // ---- pool + classify ----
    hipMemsetAsync(sums, 0, (size_t)(N_GRAPHS * NCOL + 512) * sizeof(float), stream);
    pool_accum_kernel<<<(N_NODES + 7) / 8, blk, 0, stream>>>(bufB, bat, sums, cnts, N_NODES);
    classify_kernel<<<(N_GRAPHS + 255) / 256, blk, 0, stream>>>(sums, cnts, Wc, bc, (float*)d_out);
}